// attn_47528108098062
// MI455X (gfx1250) — compile-verified
//
#include <hip/hip_runtime.h>
#include <stdint.h>

typedef float v2f __attribute__((ext_vector_type(2)));
typedef float v8f __attribute__((ext_vector_type(8)));

__device__ __forceinline__ v8f wmma_f32(v2f a, v2f b, v8f c) {
  // V_WMMA_F32_16X16X4_F32 : exact fp32, D = A(16x4) * B(4x16) + C(16x16)
  return __builtin_amdgcn_wmma_f32_16x16x4_f32(false, a, false, b, (short)0, c,
                                               false, false);
}

// CDNA5 async global->LDS copy (ASYNCcnt-tracked, bypasses VGPRs).
// GVS addressing: mem = SADDR(64) + VADDR(i32) ; VDST = per-lane LDS byte addr.
__device__ __forceinline__ void async_copy_b128(uint32_t lds_addr,
                                                uint32_t gl_off,
                                                const void* sbase) {
  asm volatile("global_load_async_to_lds_b128 %0, %1, %2"
               :: "v"(lds_addr), "v"(gl_off), "s"(sbase)
               : "memory");
}
#define WAIT_ASYNC(imm) asm volatile("s_wait_asynccnt " #imm ::: "memory")

// LDS byte offset of a __shared__ object = low 32 bits of its aperture address
__device__ __forceinline__ uint32_t lds_off(const void* p) {
  return (uint32_t)(uintptr_t)p;
}

// ---------------------------------------------------------------------------
// Generic GEMM:  C[M,N] = A[M,K] @ B[K,N] + bias[N]   (row-major, fp32)
// Block tile 128x128, BK=16, double-buffered async LDS staging.
// 256 threads = 8 waves (2x4); each wave owns 64x32 = 4x2 C fragments.
// As kept row-major [m][k] with row pad 20 -> A frag is one aligned b64 load.
// ---------------------------------------------------------------------------
#define ASTR 20  // floats per As row (80 B: 16B-aligned, conflict-free)

__global__ __launch_bounds__(256)
void gemm_bias_wmma(const float* __restrict__ A, const float* __restrict__ B,
                    const float* __restrict__ bias, float* __restrict__ C,
                    int M, int N, int K) {
  __shared__ __align__(16) float As[2][128 * ASTR];
  __shared__ __align__(16) float Bs[2][16 * 128];

  const int tid   = threadIdx.x;
  const int lane  = tid & 31;
  const int wid   = tid >> 5;    // 0..7
  const int wm    = wid >> 2;    // 0..1 -> 64 rows
  const int wn    = wid & 3;     // 0..3 -> 32 cols
  const int m0    = blockIdx.y * 128;
  const int n0    = blockIdx.x * 128;
  const int lhalf = lane >> 4;
  const int l16   = lane & 15;
  const int kb    = lhalf * 2;   // K base within a 4-chunk (A & B identical)

  // staging tasks: A = 128 rows x 4 b128-groups, B = 16 rows x 32 b128-groups
  const int am0 = tid >> 2;           // A rows for task0 / task1
  const int am1 = am0 + 64;
  const int akg = (tid & 3) * 4;      // k group (floats)
  const int bk0 = tid >> 5;           // B k rows for task0 / task1
  const int bk1 = bk0 + 8;
  const int bng = (tid & 31) * 4;     // n group (floats)

  const uint32_t asb[2] = {lds_off(&As[0][0]), lds_off(&As[1][0])};
  const uint32_t bsb[2] = {lds_off(&Bs[0][0]), lds_off(&Bs[1][0])};

  v8f acc[4][2];
#pragma unroll
  for (int i = 0; i < 4; ++i)
#pragma unroll
    for (int j = 0; j < 2; ++j)
#pragma unroll
      for (int r = 0; r < 8; ++r) acc[i][j][r] = 0.0f;

#define GEMM_ISSUE(buf, k0)                                                    \
  do {                                                                         \
    async_copy_b128(asb[buf] + (uint32_t)(am0 * ASTR + akg) * 4u,              \
                    (uint32_t)(((m0 + am0) * K + (k0) + akg) * 4), A);         \
    async_copy_b128(asb[buf] + (uint32_t)(am1 * ASTR + akg) * 4u,              \
                    (uint32_t)(((m0 + am1) * K + (k0) + akg) * 4), A);         \
    async_copy_b128(bsb[buf] + (uint32_t)(bk0 * 128 + bng) * 4u,               \
                    (uint32_t)((((k0) + bk0) * N + n0 + bng) * 4), B);         \
    async_copy_b128(bsb[buf] + (uint32_t)(bk1 * 128 + bng) * 4u,               \
                    (uint32_t)((((k0) + bk1) * N + n0 + bng) * 4), B);         \
  } while (0)

  int cur = 0;
  GEMM_ISSUE(0, 0);

  for (int k0 = 0; k0 < K; k0 += 16) {
    if (k0 + 16 < K) {
      GEMM_ISSUE(cur ^ 1, k0 + 16);
      WAIT_ASYNC(0x4);   // drain current tile's 4 copies, keep next in flight
    } else {
      WAIT_ASYNC(0x0);
    }
    __syncthreads();

    const float* as = As[cur];
    const float* bs = Bs[cur];
#pragma unroll
    for (int kk = 0; kk < 16; kk += 4) {
      v2f afr[4], bfr[2];
#pragma unroll
      for (int i = 0; i < 4; ++i) {
        const int m = wm * 64 + i * 16 + l16;
        afr[i] = *(const v2f*)&as[m * ASTR + kk + kb];   // aligned b64
      }
#pragma unroll
      for (int j = 0; j < 2; ++j) {
        const int n = wn * 32 + j * 16 + l16;
        bfr[j][0] = bs[(kk + kb + 0) * 128 + n];
        bfr[j][1] = bs[(kk + kb + 1) * 128 + n];
      }
#pragma unroll
      for (int i = 0; i < 4; ++i)
#pragma unroll
        for (int j = 0; j < 2; ++j)
          acc[i][j] = wmma_f32(afr[i], bfr[j], acc[i][j]);
    }
    __syncthreads();  // readers done before this buffer is refilled
    cur ^= 1;
  }
#undef GEMM_ISSUE

  // Epilogue: C frag element r -> M = r + 8*lhalf, N = l16
#pragma unroll
  for (int i = 0; i < 4; ++i)
#pragma unroll
    for (int j = 0; j < 2; ++j) {
      const int n = n0 + wn * 32 + j * 16 + l16;
      const float bv = bias[n];
#pragma unroll
      for (int r = 0; r < 8; ++r) {
        const int m = m0 + wm * 64 + i * 16 + r + lhalf * 8;
        C[(size_t)m * N + n] = acc[i][j][r] + bv;
      }
    }
}

// ---------------------------------------------------------------------------
// present[b][s][h][t][d] = c[b*T+t][ (s+1)*1024 + h*64 + d ]   (float4 copy)
// ---------------------------------------------------------------------------
__global__ void pack_present(const float* __restrict__ c,
                             float* __restrict__ present, int T, int total4) {
  int idx = blockIdx.x * blockDim.x + threadIdx.x;
  if (idx >= total4) return;
  const int d4 = idx & 15;
  int rest = idx >> 4;
  const int t = rest % T; rest /= T;
  const int h = rest & 15; rest >>= 4;
  const int s = rest & 1;
  const int b = rest >> 1;
  const float4 v = *(const float4*)(c + ((size_t)(b * T + t)) * 3072 +
                                    (size_t)(s + 1) * 1024 + h * 64 + d4 * 4);
  *(float4*)(present +
             ((((size_t)(b * 2 + s) * 16 + h) * T + t) * 64 + d4 * 4)) = v;
}

// ---------------------------------------------------------------------------
// Fused causal flash attention, fp32 WMMA, async double-buffered K/V tiles.
// Grid: (T/64, H, B). Block: 128 threads = 4 waves; wave w owns q rows
// [w*16, w*16+16). Tiles in LDS, stride 68 floats (bank pad, 16B-aligned rows).
// ---------------------------------------------------------------------------
#define LDSTR 68

__global__ __launch_bounds__(128)
void attn_flash_wmma(const float* __restrict__ c, float* __restrict__ out,
                     int T) {
  const int qt    = blockIdx.x;
  const int h     = blockIdx.y;
  const int b     = blockIdx.z;
  const int tid   = threadIdx.x;
  const int lane  = tid & 31;
  const int w     = tid >> 5;
  const int lhalf = lane >> 4;
  const int l16   = lane & 15;
  const int q0    = qt * 64;

  __shared__ __align__(16) float Qs[64 * LDSTR];
  __shared__ __align__(16) float Ks[2][64 * LDSTR];
  __shared__ __align__(16) float Vs[2][64 * LDSTR];
  __shared__ __align__(16) float Ps[64 * LDSTR];

  const float* base = c + (size_t)b * T * 3072 + h * 64;  // qkv row stride 3072

  // staging tasks: 64 rows x 16 b128-groups = 1024; 8 per thread
  const int sr = tid >> 4;          // base row (0..7), rows sr + 8*i
  const int sd = (tid & 15) * 4;    // float offset within 64-wide row
  const uint32_t qsb = lds_off(Qs);
  const uint32_t ksb[2] = {lds_off(&Ks[0][0]), lds_off(&Ks[1][0])};
  const uint32_t vsb[2] = {lds_off(&Vs[0][0]), lds_off(&Vs[1][0])};

  // Q tile: 8 async b128 per thread
#pragma unroll
  for (int i = 0; i < 8; ++i) {
    const int r = sr + 8 * i;
    async_copy_b128(qsb + (uint32_t)(r * LDSTR + sd) * 4u,
                    (uint32_t)(((q0 + r) * 3072 + sd) * 4), base);
  }

#define KV_ISSUE(buf, k0)                                                      \
  do {                                                                         \
    _Pragma("unroll") for (int i = 0; i < 8; ++i) {                            \
      const int r = sr + 8 * i;                                                \
      async_copy_b128(ksb[buf] + (uint32_t)(r * LDSTR + sd) * 4u,              \
                      (uint32_t)((((k0) + r) * 3072 + 1024 + sd) * 4), base);  \
      async_copy_b128(vsb[buf] + (uint32_t)(r * LDSTR + sd) * 4u,              \
                      (uint32_t)((((k0) + r) * 3072 + 2048 + sd) * 4), base);  \
    }                                                                          \
  } while (0)

  v8f oacc[4];
  float mrow[8], lrow[8];
#pragma unroll
  for (int nt = 0; nt < 4; ++nt)
#pragma unroll
    for (int r = 0; r < 8; ++r) oacc[nt][r] = 0.0f;
#pragma unroll
  for (int r = 0; r < 8; ++r) { mrow[r] = -3.0e38f; lrow[r] = 0.0f; }

  const float scale = 0.125f;  // rsqrt(hd=64)

  int cur = 0;
  KV_ISSUE(0, 0);

  for (int kt = 0; kt <= qt; ++kt) {
    const int k0 = kt * 64;
    if (kt < qt) {
      KV_ISSUE(cur ^ 1, k0 + 64);
      WAIT_ASYNC(0x10);  // drain Q + current K/V, keep next tile in flight
    } else {
      WAIT_ASYNC(0x0);
    }
    __syncthreads();

    const float* ks = Ks[cur];
    const float* vs = Vs[cur];

    // S = Q @ K^T  (per-wave 16x64 in 4 fragments)
    v8f s[4];
#pragma unroll
    for (int nt = 0; nt < 4; ++nt) {
      v8f a8{};
#pragma unroll
      for (int kd = 0; kd < 64; kd += 4) {
        const v2f av = *(const v2f*)&Qs[(w * 16 + l16) * LDSTR + kd + lhalf * 2];
        const v2f bv = *(const v2f*)&ks[(nt * 16 + l16) * LDSTR + kd + lhalf * 2];
        a8 = wmma_f32(av, bv, a8);
      }
      s[nt] = a8;
    }

    // scale + causal mask (-1e9, faithful to reference), per-row max
    float rmax[8];
#pragma unroll
    for (int r = 0; r < 8; ++r) rmax[r] = -3.0e38f;
#pragma unroll
    for (int nt = 0; nt < 4; ++nt)
#pragma unroll
      for (int r = 0; r < 8; ++r) {
        const int qrow = q0 + w * 16 + r + lhalf * 8;
        const int kcol = k0 + nt * 16 + l16;
        float v = s[nt][r] * scale;
        v = (kcol <= qrow) ? v : -1.0e9f;
        s[nt][r] = v;
        rmax[r] = fmaxf(rmax[r], v);
      }
#pragma unroll
    for (int off = 8; off >= 1; off >>= 1)
#pragma unroll
      for (int r = 0; r < 8; ++r)
        rmax[r] = fmaxf(rmax[r], __shfl_xor(rmax[r], off, 32));

    // online softmax update
    float pscale[8];
#pragma unroll
    for (int r = 0; r < 8; ++r) {
      const float mnew = fmaxf(mrow[r], rmax[r]);
      pscale[r] = __expf(mrow[r] - mnew);
      mrow[r] = mnew;
    }
    float rsum[8];
#pragma unroll
    for (int r = 0; r < 8; ++r) rsum[r] = 0.0f;
#pragma unroll
    for (int nt = 0; nt < 4; ++nt)
#pragma unroll
      for (int r = 0; r < 8; ++r) {
        const float p = __expf(s[nt][r] - mrow[r]);
        s[nt][r] = p;
        rsum[r] += p;
      }
#pragma unroll
    for (int off = 8; off >= 1; off >>= 1)
#pragma unroll
      for (int r = 0; r < 8; ++r) rsum[r] += __shfl_xor(rsum[r], off, 32);
#pragma unroll
    for (int r = 0; r < 8; ++r) lrow[r] = lrow[r] * pscale[r] + rsum[r];
#pragma unroll
    for (int nt = 0; nt < 4; ++nt)
#pragma unroll
      for (int r = 0; r < 8; ++r) oacc[nt][r] *= pscale[r];

    // C-layout P -> LDS (wave-private rows), reload as A fragments
#pragma unroll
    for (int nt = 0; nt < 4; ++nt)
#pragma unroll
      for (int r = 0; r < 8; ++r)
        Ps[(w * 16 + r + lhalf * 8) * LDSTR + nt * 16 + l16] = s[nt][r];
    // same-wave LDS ops are in order: no barrier needed for wave-private P

    // O += P @ V
#pragma unroll
    for (int nt = 0; nt < 4; ++nt) {
      v8f a8 = oacc[nt];
#pragma unroll
      for (int kk = 0; kk < 64; kk += 4) {
        const v2f av = *(const v2f*)&Ps[(w * 16 + l16) * LDSTR + kk + lhalf * 2];
        v2f bv;
        bv[0] = vs[(kk + lhalf * 2 + 0) * LDSTR + nt * 16 + l16];
        bv[1] = vs[(kk + lhalf * 2 + 1) * LDSTR + nt * 16 + l16];
        a8 = wmma_f32(av, bv, a8);
      }
      oacc[nt] = a8;
    }

    __syncthreads();  // readers done before buffer refill next iteration
    cur ^= 1;
  }
#undef KV_ISSUE

  // normalize and write head-merged output
#pragma unroll
  for (int r = 0; r < 8; ++r) {
    const float inv = 1.0f / lrow[r];
    const int m = q0 + w * 16 + r + lhalf * 8;
#pragma unroll
    for (int nt = 0; nt < 4; ++nt)
      out[((size_t)b * T + m) * 1024 + h * 64 + nt * 16 + l16] =
          oacc[nt][r] * inv;
  }
}

// ---------------------------------------------------------------------------
extern "C" void kernel_launch(void* const* d_in, const int* in_sizes, int n_in,
                              void* d_out, int out_size, void* d_ws,
                              size_t ws_size, hipStream_t stream) {
  (void)in_sizes; (void)n_in; (void)out_size; (void)ws_size;
  const float* x      = (const float*)d_in[0];  // [2,2048,1024]
  const float* w_attn = (const float*)d_in[1];  // [1024,3072]
  const float* b_attn = (const float*)d_in[2];  // [3072]
  const float* w_proj = (const float*)d_in[3];  // [1024,1024]
  const float* b_proj = (const float*)d_in[4];  // [1024]

  const int B = 2, T = 2048, E = 1024, M = B * T;  // M = 4096

  float* out_a       = (float*)d_out;             // [B,T,E]
  float* out_present = out_a + (size_t)M * E;     // [B,2,H,T,hd]

  float* cbuf = (float*)d_ws;                     // [M, 3E] = 48 MB
  float* abuf = cbuf + (size_t)M * 3 * E;         // [M, E]  = 16 MB

  // 1) QKV projection: c = x @ w_attn + b_attn
  gemm_bias_wmma<<<dim3(3 * E / 128, M / 128), dim3(256), 0, stream>>>(
      x, w_attn, b_attn, cbuf, M, 3 * E, E);

  // 2) present = stack(k, v)
  const int total4 = B * 2 * 16 * T * 16;
  pack_present<<<(total4 + 255) / 256, 256, 0, stream>>>(cbuf, out_present, T,
                                                         total4);

  // 3) fused causal attention, head-merged output
  attn_flash_wmma<<<dim3(T / 64, 16, B), dim3(128), 0, stream>>>(cbuf, abuf, T);

  // 4) output projection: a = attn @ w_proj + b_proj
  gemm_bias_wmma<<<dim3(E / 128, M / 128), dim3(256), 0, stream>>>(
      abuf, w_proj, b_proj, out_a, M, E, E);
}